// self_attention_56667798503751
// MI455X (gfx1250) — compile-verified
//
#include <hip/hip_runtime.h>
#include <math.h>

// Problem constants (match reference)
#define BB 4
#define SS 2048
#define CC 512
#define HH 8
#define DD 64
#define HD 512   // H*D == C

typedef __attribute__((ext_vector_type(2))) float v2f;
typedef __attribute__((ext_vector_type(8))) float v8f;

__device__ __forceinline__ v8f wmma_f32(v2f a, v2f b, v8f c) {
  // 8 args: (neg_a, A, neg_b, B, c_mod, C, reuse_a, reuse_b)
  return __builtin_amdgcn_wmma_f32_16x16x4_f32(false, a, false, b, (short)0, c,
                                               false, false);
}

// WGP-scope prefetch (scope 0: pull into all cache levels; data is already
// L2-resident so DEV-scope prefetch would be a no-op).
__device__ __forceinline__ void prefetch_wgp(const void* p) {
  asm volatile("global_prefetch_b8 %0, off" ::"v"(p));
}

// ---------------------------------------------------------------------------
// GEMM: Y[8192 x 512] = X[8192 x 512] * W[512 x 512], fp32 row-major.
// Block = 128 threads (4 waves). Each wave owns a 32(M) x 64(N) strip
// (two 16-row WMMA tiles sharing the B fragments -> 10 VMEM per 8 WMMA).
// gridDim = (64, 8): 256 32-row strips / 4 waves, 8 N-strips of 64.
// ---------------------------------------------------------------------------
__global__ __launch_bounds__(128) void gemm_8192x512x512(
    const float* __restrict__ X, const float* __restrict__ W,
    float* __restrict__ Y) {
  const int lane = threadIdx.x & 31;
  const int wave = threadIdx.x >> 5;
  const int mbase = (blockIdx.x * 4 + wave) * 32;
  const int nbase = blockIdx.y * 64;
  const int lr = lane & 15;          // M row (A) / N col (B,C)
  const int kh = (lane >> 4) * 2;    // K offset for upper half-wave

  const float* xrow0 = X + (size_t)(mbase + lr) * HD;
  const float* xrow1 = xrow0 + (size_t)16 * HD;

  v8f acc[8] = {};  // acc[t*4 + j]: t = M sub-tile (0/1), j = 16-col group

  for (int c0 = 0; c0 < CC; c0 += 16) {
    // Prefetch the next 16-K block into the WGP caches with per-lane spread
    // addresses: X -> 32 rows (1 line each); W -> 16 rows x 4 column-lines.
    if (c0 + 16 < CC) {
      prefetch_wgp(X + (size_t)(mbase + lane) * HD + c0 + 16);
      const float* wpf =
          W + (size_t)(c0 + 16 + lr) * HD + nbase + ((lane >> 4) << 5);
      prefetch_wgp(wpf);
      prefetch_wgp(wpf + 16);
    }
#pragma unroll
    for (int u = 0; u < 4; ++u) {
      const int c = c0 + 4 * u;
      // A fragments: lanes 0-15 hold (M=lr, K=c+{0,1}); lanes 16-31 K=c+{2,3}
      v2f a0 = *(const v2f*)(xrow0 + c + kh);
      v2f a1 = *(const v2f*)(xrow1 + c + kh);
      // B fragments: lane = N; VGPR pair = consecutive K within half
      const float* wp = W + (size_t)(c + kh) * HD + nbase + lr;
      v2f b0, b1, b2, b3;
      b0.x = wp[0];   b0.y = wp[HD + 0];
      b1.x = wp[16];  b1.y = wp[HD + 16];
      b2.x = wp[32];  b2.y = wp[HD + 32];
      b3.x = wp[48];  b3.y = wp[HD + 48];
      acc[0] = wmma_f32(a0, b0, acc[0]);
      acc[1] = wmma_f32(a0, b1, acc[1]);
      acc[2] = wmma_f32(a0, b2, acc[2]);
      acc[3] = wmma_f32(a0, b3, acc[3]);
      acc[4] = wmma_f32(a1, b0, acc[4]);
      acc[5] = wmma_f32(a1, b1, acc[5]);
      acc[6] = wmma_f32(a1, b2, acc[6]);
      acc[7] = wmma_f32(a1, b3, acc[7]);
    }
  }

  // C layout: VGPR r -> row r (lanes 0-15) / row r+8 (lanes 16-31), col = lane%16
#pragma unroll
  for (int t = 0; t < 2; ++t) {
    float* yb =
        Y + (size_t)(mbase + t * 16 + 8 * (lane >> 4)) * HD + nbase + lr;
#pragma unroll
    for (int r = 0; r < 8; ++r) {
      float* yr = yb + (size_t)r * HD;
      yr[0]  = acc[t * 4 + 0][r];
      yr[16] = acc[t * 4 + 1][r];
      yr[32] = acc[t * 4 + 2][r];
      yr[48] = acc[t * 4 + 3][r];
    }
  }
}

// ---------------------------------------------------------------------------
// Attention: one wave (32 threads) per (b, h, 16-query tile).
// Flash-style online softmax over key tiles of 16, keys >= length masked.
// Output O[b, s, h*64 + d], rows with q >= length zeroed (matches reference).
// ---------------------------------------------------------------------------
__global__ __launch_bounds__(32) void attn_kernel(
    const float* __restrict__ Q, const float* __restrict__ K,
    const float* __restrict__ V, const int* __restrict__ length,
    float* __restrict__ O) {
  __shared__ float lds[16 * 16];

  const int lane  = threadIdx.x & 31;
  const int lr    = lane & 15;
  const int kh    = (lane >> 4) * 2;
  const int half8 = (lane >> 4) * 8;
  const int qbase = blockIdx.x * 16;
  const int h     = blockIdx.y;
  const int b     = blockIdx.z;

  int L = length[b];
  if (L < 0) L = 0;
  if (L > SS) L = SS;

  const size_t baseBH = (size_t)b * SS * HD + (size_t)h * DD;
  const float* Qb = Q + baseBH;
  const float* Kb = K + baseBH;
  const float* Vb = V + baseBH;
  float*       Ob = O + baseBH;

  // Preload Q A-fragments for the 16 K-steps across D=64
  v2f aq[16];
  {
    const float* qrow = Qb + (size_t)(qbase + lr) * HD;
#pragma unroll
    for (int s = 0; s < 16; ++s) aq[s] = *(const v2f*)(qrow + 4 * s + kh);
  }

  float m[8], l[8];
#pragma unroll
  for (int r = 0; r < 8; ++r) { m[r] = -3.0e38f; l[r] = 0.0f; }
  v8f acc[4] = {};

  const int nkt = (L + 15) >> 4;
  for (int kt = 0; kt < nkt; ++kt) {
    const int kbase = kt * 16;

    // Prefetch next key tile's K and V rows (WGP scope, lanes spread over
    // 16 rows x 2 of 4 column-lines; alternate column pair by tile parity).
    if (kt + 1 < nkt) {
      const size_t poff = (size_t)(kbase + 16 + lr) * HD +
                          ((lane >> 4) << 5) + ((kt & 1) << 4);
      prefetch_wgp(Kb + poff);
      prefetch_wgp(Vb + poff);
    }

    // scores tile S = (Q_tile)(K_tile)^T, K-loop over D=64
    v8f sc = {};
    const float* krow = Kb + (size_t)(kbase + lr) * HD;
#pragma unroll
    for (int s = 0; s < 16; ++s) {
      v2f bk = *(const v2f*)(krow + 4 * s + kh);
      sc = wmma_f32(aq[s], bk, sc);
    }

    const bool kvalid = (kbase + lr) < L;
#pragma unroll
    for (int r = 0; r < 8; ++r) {
      float v = sc[r] * 0.125f;  // 1/sqrt(64)
      sc[r] = kvalid ? v : -1.0e30f;
    }

    // online softmax (per row; 16-lane group = key dimension)
    float p[8], scale[8];
#pragma unroll
    for (int r = 0; r < 8; ++r) {
      float rowmax = sc[r];
      rowmax = fmaxf(rowmax, __shfl_xor(rowmax, 1));
      rowmax = fmaxf(rowmax, __shfl_xor(rowmax, 2));
      rowmax = fmaxf(rowmax, __shfl_xor(rowmax, 4));
      rowmax = fmaxf(rowmax, __shfl_xor(rowmax, 8));
      const float mnew = fmaxf(m[r], rowmax);
      const float pr = __expf(sc[r] - mnew);
      float rowsum = pr;
      rowsum += __shfl_xor(rowsum, 1);
      rowsum += __shfl_xor(rowsum, 2);
      rowsum += __shfl_xor(rowsum, 4);
      rowsum += __shfl_xor(rowsum, 8);
      scale[r] = __expf(m[r] - mnew);
      l[r] = l[r] * scale[r] + rowsum;
      m[r] = mnew;
      p[r] = pr;
    }
#pragma unroll
    for (int j = 0; j < 4; ++j)
#pragma unroll
      for (int r = 0; r < 8; ++r) acc[j][r] *= scale[r];

    // relayout P: C-layout (row across lanes) -> A-layout (row within lane)
    __syncthreads();
#pragma unroll
    for (int r = 0; r < 8; ++r) lds[(r + half8) * 16 + lr] = p[r];
    __syncthreads();

    // O_tile += P(16x16) x V_tile(16x64); K-dim = 16 keys -> 4 steps
#pragma unroll
    for (int s = 0; s < 4; ++s) {
      v2f ap;
      ap.x = lds[lr * 16 + 4 * s + kh];
      ap.y = lds[lr * 16 + 4 * s + kh + 1];
      const float* vrow = Vb + (size_t)(kbase + 4 * s + kh) * HD;
#pragma unroll
      for (int j = 0; j < 4; ++j) {
        v2f bv;
        bv.x = vrow[j * 16 + lr];
        bv.y = vrow[HD + j * 16 + lr];
        acc[j] = wmma_f32(ap, bv, acc[j]);
      }
    }
  }

  // finalize: normalize by row sums; zero rows beyond length
#pragma unroll
  for (int r = 0; r < 8; ++r) {
    const int qrow = qbase + r + half8;
    const float inv = (qrow < L && l[r] > 0.0f) ? (1.0f / l[r]) : 0.0f;
    float* orow = Ob + (size_t)qrow * HD;
    orow[0  + lr] = acc[0][r] * inv;
    orow[16 + lr] = acc[1][r] * inv;
    orow[32 + lr] = acc[2][r] * inv;
    orow[48 + lr] = acc[3][r] * inv;
  }
}

// ---------------------------------------------------------------------------
extern "C" void kernel_launch(void* const* d_in, const int* in_sizes, int n_in,
                              void* d_out, int out_size, void* d_ws,
                              size_t ws_size, hipStream_t stream) {
  const float* x      = (const float*)d_in[0];
  const int*   length = (const int*)d_in[1];
  const float* Wq     = (const float*)d_in[2];
  const float* Wk     = (const float*)d_in[3];
  const float* Wv     = (const float*)d_in[4];
  const float* Wx     = (const float*)d_in[5];
  float*       out    = (float*)d_out;

  const size_t mat = (size_t)BB * SS * HD;  // 4,194,304 floats = 16 MB
  float* Qw = (float*)d_ws;
  float* Kw = Qw + mat;
  float* Vw = Kw + mat;
  float* Ow = Vw + mat;

  const dim3 gGrid(64, 8, 1);    // 256 32-row strips / 4 waves, 8 N-strips
  const dim3 gBlock(128, 1, 1);

  gemm_8192x512x512<<<gGrid, gBlock, 0, stream>>>(x, Wq, Qw);
  gemm_8192x512x512<<<gGrid, gBlock, 0, stream>>>(x, Wk, Kw);
  gemm_8192x512x512<<<gGrid, gBlock, 0, stream>>>(x, Wv, Vw);

  attn_kernel<<<dim3(SS / 16, HH, BB), dim3(32, 1, 1), 0, stream>>>(
      Qw, Kw, Vw, length, Ow);

  gemm_8192x512x512<<<gGrid, gBlock, 0, stream>>>(Ow, Wx, out);
}